// GCNEncoder_54924041781475
// MI455X (gfx1250) — compile-verified
//
#include <hip/hip_runtime.h>
#include <hip/hip_bf16.h>

typedef __attribute__((ext_vector_type(2))) float v2f;
typedef __attribute__((ext_vector_type(8))) float v8f;

#define D_IN  128
#define D_HID 128
#define D_OUT 64

// ---------------- degree / symmetric norm ----------------
__global__ void k_deg_init(float* __restrict__ deg, int n) {
  int i = blockIdx.x * blockDim.x + threadIdx.x;
  if (i < n) deg[i] = 1.0f;                 // self-loop contributes 1
}

__global__ void k_deg_count(const int* __restrict__ dst, float* __restrict__ deg, int e) {
  int i = blockIdx.x * blockDim.x + threadIdx.x;
  if (i < e) atomicAdd(&deg[dst[i]], 1.0f);
}

__global__ void k_deg_rsqrt(float* __restrict__ deg, int n) {
  int i = blockIdx.x * blockDim.x + threadIdx.x;
  if (i < n) {
    float d = deg[i];
    deg[i] = (d > 0.0f) ? rsqrtf(d) : 0.0f; // -> dinv
  }
}

// ---------------- dense GEMM: Y[N,NC] = act(X) @ W[K,NC] ----------------
// K, NC compile-time so the fully-unrolled loop uses immediate-offset
// global_load + v_wmma_f32_16x16x4_f32 only (no per-iter 64-bit addr math).
// FUSE_IN: A element = relu(X[i][k] + BIN[k])  (fuses previous layer's bias+ReLU)
// One wave32 computes one 16x16 f32 output tile.
// A 16x4 layout: lanes 0-15 hold M=lane, K=k0+{0,1}; lanes 16-31 M=lane-16, K=k0+{2,3}.
// B 4x16 mirrors A's K striping; N = lane&15.  C/D: VGPR i -> (M = i | i+8, N = lane&15).
template <int K, int NC, bool FUSE_IN>
__global__ void k_gemm_wmma(const float* __restrict__ X, const float* __restrict__ BIN,
                            const float* __restrict__ W, float* __restrict__ Y, int N) {
  const int lane = threadIdx.x & 31;
  const int wave = threadIdx.x >> 5;
  const int row0 = blockIdx.x << 4;
  const int col0 = ((blockIdx.y * (blockDim.x >> 5)) + wave) << 4;
  const int m    = lane & 15;             // A row within tile == B/C col within tile
  const int koff = (lane >> 4) << 1;      // 0 or 2

  const int arow_idx = min(row0 + m, N - 1);   // clamp (N is a multiple of 16 anyway)
  const float* __restrict__ arow = X + (size_t)arow_idx * K + koff;
  const float* __restrict__ bcol = W + (size_t)koff * NC + col0 + m;
  const float* __restrict__ brow = FUSE_IN ? (BIN + koff) : nullptr;

  v8f acc = {};
#pragma unroll
  for (int k0 = 0; k0 < K; k0 += 4) {
    v2f a, b;
    float a0 = arow[k0];
    float a1 = arow[k0 + 1];
    if (FUSE_IN) {                         // h = relu(agg_prev + bias_prev)
      a0 = fmaxf(a0 + brow[k0], 0.0f);
      a1 = fmaxf(a1 + brow[k0 + 1], 0.0f);
    }
    a.x = a0;
    a.y = a1;
    b.x = bcol[(size_t)k0 * NC];           // constant offsets after unroll
    b.y = bcol[(size_t)k0 * NC + NC];
    acc = __builtin_amdgcn_wmma_f32_16x16x4_f32(false, a, false, b, (short)0, acc,
                                                false, false);
  }

  const int mb = (lane >> 4) << 3;        // 0 or 8
  float* yp = Y + (size_t)(row0 + mb) * NC + col0 + m;
#pragma unroll
  for (int i = 0; i < 8; ++i) {
    if (row0 + mb + i < N) yp[(size_t)i * NC] = acc[i];
  }
}

// ---------------- agg init with self-loop term: agg = xw * dinv^2 ----------------
__global__ void k_self_init(const float* __restrict__ xw, const float* __restrict__ dinv,
                            float* __restrict__ agg, int total, int fshift) {
  int i = blockIdx.x * blockDim.x + threadIdx.x;
  if (i < total) {
    float di = dinv[i >> fshift];
    agg[i] = xw[i] * di * di;
  }
}

// ---------------- per-edge gather/scale/scatter-add (one wave32 per edge) --------
template <int PER>   // PER = F/32 floats per lane (4 for F=128, 2 for F=64)
__global__ void k_edge_scatter(const float* __restrict__ xw, float* __restrict__ agg,
                               const int* __restrict__ src, const int* __restrict__ dst,
                               const float* __restrict__ dinv, int n_edges) {
  const int e = blockIdx.x * (blockDim.x >> 5) + (threadIdx.x >> 5);
  if (e >= n_edges) return;
  const int lane = threadIdx.x & 31;
  const int s = src[e];
  const int d = dst[e];
  const float norm = dinv[s] * dinv[d];
  const int F = PER * 32;
  const float* __restrict__ xs = xw + (size_t)s * F + lane * PER;
  float* __restrict__ ad = agg + (size_t)d * F + lane * PER;
  if (PER == 4) {
    float4 v = *(const float4*)xs;        // global_load_b128
    atomicAdd(&ad[0], v.x * norm);        // global_atomic_add_f32 (no return)
    atomicAdd(&ad[1], v.y * norm);
    atomicAdd(&ad[2], v.z * norm);
    atomicAdd(&ad[3], v.w * norm);
  } else {
    float2 v = *(const float2*)xs;        // global_load_b64
    atomicAdd(&ad[0], v.x * norm);
    atomicAdd(&ad[1], v.y * norm);
  }
}

// ---------------- final bias, in place on d_out ----------------
__global__ void k_bias(float* __restrict__ h, const float* __restrict__ b,
                       int total, int fmask) {
  int i = blockIdx.x * blockDim.x + threadIdx.x;
  if (i < total) h[i] = h[i] + b[i & fmask];
}

extern "C" void kernel_launch(void* const* d_in, const int* in_sizes, int n_in,
                              void* d_out, int out_size, void* d_ws, size_t ws_size,
                              hipStream_t stream) {
  const float* x  = (const float*)d_in[0];
  const int*   ei = (const int*)d_in[1];
  const float* W1 = (const float*)d_in[2];
  const float* b1 = (const float*)d_in[3];
  const float* W2 = (const float*)d_in[4];
  const float* b2 = (const float*)d_in[5];
  const float* W3 = (const float*)d_in[6];
  const float* b3 = (const float*)d_in[7];
  float* out = (float*)d_out;

  const int N = in_sizes[0] / D_IN;       // 100000
  const int E = in_sizes[1] / 2;          // 1600000
  const int* src = ei;                    // edge_index row 0
  const int* dst = ei + E;                // edge_index row 1

  // workspace layout: dinv | bufA (N*128) | bufB (N*128)   (~103 MB)
  float* dinv = (float*)d_ws;
  float* bufA = dinv + (((size_t)N + 255) & ~(size_t)255);
  float* bufB = bufA + (size_t)N * D_HID;

  const int T = 256;
  // ---- normalization: deg -> dinv ----
  k_deg_init <<<(N + T - 1) / T, T, 0, stream>>>(dinv, N);
  k_deg_count<<<(E + T - 1) / T, T, 0, stream>>>(dst, dinv, E);
  k_deg_rsqrt<<<(N + T - 1) / T, T, 0, stream>>>(dinv, N);

  const int rowTiles = (N + 15) / 16;     // 6250
  const dim3 gemmBlk(128);                // 4 waves -> 4 col tiles (64 cols) per block

  // ---- layer 1: xw1 = x@W1 -> bufA; agg1 (raw, pre-bias/ReLU) -> bufB ----
  k_gemm_wmma<D_IN, D_HID, false>
      <<<dim3(rowTiles, D_HID / 64), gemmBlk, 0, stream>>>(x, nullptr, W1, bufA, N);
  {
    int total = N * D_HID;
    k_self_init<<<(total + T - 1) / T, T, 0, stream>>>(bufA, dinv, bufB, total, 7);
    k_edge_scatter<4><<<(E + 7) / 8, 256, 0, stream>>>(bufA, bufB, src, dst, dinv, E);
  }

  // ---- layer 2: xw2 = relu(agg1+b1)@W2 -> bufA; agg2 (raw) -> bufB ----
  k_gemm_wmma<D_HID, D_HID, true>
      <<<dim3(rowTiles, D_HID / 64), gemmBlk, 0, stream>>>(bufB, b1, W2, bufA, N);
  {
    int total = N * D_HID;
    k_self_init<<<(total + T - 1) / T, T, 0, stream>>>(bufA, dinv, bufB, total, 7);
    k_edge_scatter<4><<<(E + 7) / 8, 256, 0, stream>>>(bufA, bufB, src, dst, dinv, E);
  }

  // ---- layer 3: xw3 = relu(agg2+b2)@W3 -> bufA (N x 64); aggregate into d_out ----
  k_gemm_wmma<D_HID, D_OUT, true>
      <<<dim3(rowTiles, D_OUT / 64), gemmBlk, 0, stream>>>(bufB, b2, W3, bufA, N);
  {
    int total = N * D_OUT;
    k_self_init<<<(total + T - 1) / T, T, 0, stream>>>(bufA, dinv, out, total, 6);
    k_edge_scatter<2><<<(E + 7) / 8, 256, 0, stream>>>(bufA, out, src, dst, dinv, E);
    k_bias<<<(total + T - 1) / T, T, 0, stream>>>(out, b3, total, D_OUT - 1);
  }
}